// winAttn_22282290331768
// MI455X (gfx1250) — compile-verified
//
#include <hip/hip_runtime.h>
#include <math.h>

#define B_   8
#define S_   2048
#define E_   512
#define WIN_ 11
#define F_   (WIN_ * E_)        /* 5632 */
#define WW_  5                  /* WIN/2 */
#define KT_  256                /* K-tile (floats) per TDM transfer   */
#define NT_  (F_ / KT_)         /* 22 tiles                            */
#define FP_  (F_ + 4)           /* padded window row stride: 4 mod 64  */
#define WTS_ (KT_ + 4)          /* padded W-tile row stride: 260       */

/* LDS layout (floats) */
#define L_FLAT 0                         /* 8 * 5636                   */
#define L_WT0  (B_ * FP_)                /* 45088                      */
#define L_WT1  (L_WT0 + WIN_ * WTS_)     /* + 2860                     */
#define L_GATE (L_WT1 + WIN_ * WTS_)     /* + 2860                     */
#define L_G2   (L_GATE + 256)
#define L_TOT  (L_G2 + 96)               /* 51160 floats = 204,640 B   */

typedef __attribute__((ext_vector_type(2))) float v2f;
typedef __attribute__((ext_vector_type(8))) float v8f;
typedef __attribute__((ext_vector_type(4))) unsigned int v4u;
typedef __attribute__((ext_vector_type(4))) int v4i;
typedef __attribute__((ext_vector_type(8))) int v8i;

#if __has_include(<hip/amd_detail/amd_gfx1250_TDM.h>)
#define TDM_6ARG 1
#endif

/* Issue one TDM 2D tile load: W[s][0..10][t*256 .. +255] -> LDS (row stride 260) */
__device__ __forceinline__ void tdm_load_wtile(const float* Wbase, int s, int t,
                                               unsigned lds_byte_off)
{
    unsigned long long ga = (unsigned long long)(uintptr_t)
        (Wbase + ((size_t)s * WIN_ * F_ + (size_t)t * KT_));

    v4u g0;
    g0.x = 1u;                                   /* count=1, user load descriptor */
    g0.y = lds_byte_off;                         /* lds_addr (bytes)              */
    g0.z = (unsigned)(ga & 0xFFFFFFFFu);         /* global_addr[31:0]             */
    g0.w = (unsigned)((ga >> 32) & 0x01FFFFFFu)  /* global_addr[56:32]            */
         | (2u << 30);                           /* type = 2 (image/tensor)       */

    v8i g1;
    g1[0] = (int)((2u << 16)        /* data_size = 4B                          */
                | (1u << 20)        /* pad_enable                              */
                | (7u << 22)        /* pad_interval code 7 = every 256 DWORDs  */
                | (3u << 25));      /* pad_amount  code 3 = 4 DWORDs           */
    g1[1] = (int)((F_ & 0xFFFF) << 16);                 /* tensor_dim0 lo      */
    g1[2] = (int)(((F_ >> 16) & 0xFFFF) | (WIN_ << 16));/* dim0 hi | dim1 lo   */
    g1[3] = (int)(KT_ << 16);                           /* dim1 hi=0 | tile_dim0 */
    g1[4] = WIN_;                                       /* tile_dim1=11, tile_dim2=0 */
    g1[5] = F_;                                         /* tensor_dim0_stride lo */
    g1[6] = 0;                                          /* stride hi | dim1_stride */
    g1[7] = 0;

    v4i gz = {0, 0, 0, 0};
#ifdef TDM_6ARG
    v8i gz8 = {0, 0, 0, 0, 0, 0, 0, 0};
    __builtin_amdgcn_tensor_load_to_lds(g0, g1, gz, gz, gz8, 0);
#else
    __builtin_amdgcn_tensor_load_to_lds(g0, g1, gz, gz, 0);
#endif
}

__global__ __launch_bounds__(256)
void winattn_wmma_kernel(const float* __restrict__ x,
                         const float* __restrict__ W,
                         const float* __restrict__ bias,
                         float* __restrict__ out)
{
    __shared__ __align__(16) float lds[L_TOT];

    const int s    = blockIdx.x;
    const int tid  = threadIdx.x;
    const int lane = tid & 31;
    const int wave = tid >> 5;

    /* 32-bit LDS byte offsets (flat-address low bits == wave LDS offset) for D#. */
    const unsigned wt_off0 = (unsigned)(uintptr_t)&lds[L_WT0];
    const unsigned wt_off1 = (unsigned)(uintptr_t)&lds[L_WT1];

    /* Kick off the first two W tiles via TDM so the DMA overlaps x staging. */
    if (wave == 0) {
        tdm_load_wtile(W, s, 0, wt_off0);
        tdm_load_wtile(W, s, 1, wt_off1);
    }

    lds[L_GATE + tid] = 0.0f;       /* 256 gate accumulators */

    /* ---- Stage padded windows (8 x 5632 f32, row stride 5636) coalesced ---- */
    for (int b = 0; b < B_; ++b) {
        for (int f = tid; f < F_; f += 256) {     /* uniform trip count */
            const int w    = f >> 9;
            const int e    = f & (E_ - 1);
            const int srow = s + w - WW_;
            float v = 0.0f;
            if (srow >= 0 && srow < S_)
                v = x[((size_t)b * S_ + (size_t)srow) * E_ + e];
            lds[L_FLAT + b * FP_ + f] = v;
        }
    }
    __syncthreads();

    /* ---- Gate GEMM: M=batch(8/16), N=tap(11/16), K=5632 via f32 WMMA ---- */
    const int m    = lane & 15;
    const int half = lane >> 4;
    const int nw   = (m < WIN_) ? m : (WIN_ - 1);   /* cols 11..15 unused */
    const int mb   = m & 7;                          /* rows 8..15 unused */

    const float* aptr  = &lds[L_FLAT + mb * FP_ + wave * 4 + half * 2];
    const int    bofs  = nw * WTS_ + wave * 4 + half * 2;
    const float* wbuf0 = &lds[L_WT0 + bofs];        /* proper shared ptrs -> ds_load */
    const float* wbuf1 = &lds[L_WT1 + bofs];

    v8f acc = {};
    for (int t = 0; t < NT_; ++t) {
        if (wave == 0) {                 /* tile t guaranteed resident */
            if (t < NT_ - 1) __builtin_amdgcn_s_wait_tensorcnt(1);
            else             __builtin_amdgcn_s_wait_tensorcnt(0);
        }
        __syncthreads();

        const float* bptr = (t & 1) ? wbuf1 : wbuf0;
        #pragma unroll
        for (int j = 0; j < KT_ / 32; ++j) {         /* 8 WMMA per tile */
            v2f av = *(const v2f*)aptr;              /* ds_load, banks 4*mb+k */
            v2f bv = *(const v2f*)(bptr + 32 * j);   /* ds_load, banks 4*nw+k */
            acc = __builtin_amdgcn_wmma_f32_16x16x4_f32(
                      false, av, false, bv, (short)0, acc, false, false);
            aptr += 32;
        }
        __syncthreads();                 /* everyone done with buf (t&1) */

        if (wave == 0 && t < NT_ - 2)    /* refill it with tile t+2 */
            tdm_load_wtile(W, s, t + 2, (t & 1) ? wt_off1 : wt_off0);
    }

    /* ---- Cross-wave reduction of 16x16 partial C via ds_add_f32 ---- */
    #pragma unroll
    for (int j = 0; j < 8; ++j) {
        const int mm = j + half * 8;     /* C VGPR j: rows j / j+8 per lane half */
        atomicAdd(&lds[L_GATE + mm * 16 + m], acc[j]);
    }
    __syncthreads();

    /* ---- Bias + sigmoid for the 8x11 live gates ---- */
    if (tid < B_ * WIN_) {
        const int b = tid / WIN_;
        const int w = tid - b * WIN_;
        const float g = lds[L_GATE + b * 16 + w] + bias[s * WIN_ + w];
        lds[L_G2 + tid] = 1.0f / (1.0f + __expf(-g));
    }
    __syncthreads();

    /* ---- score[b][e] = sum_w gate * window;  out = tanh ---- */
    for (int idx = tid; idx < B_ * E_; idx += 256) {
        const int b = idx >> 9;
        const int e = idx & (E_ - 1);
        const float* wf = &lds[L_FLAT + b * FP_ + e];
        const float* gg = &lds[L_G2 + b * WIN_];
        float sum = 0.0f;
        #pragma unroll
        for (int w = 0; w < WIN_; ++w)
            sum += gg[w] * wf[w * E_];
        out[((size_t)b * S_ + (size_t)s) * E_ + e] = tanhf(sum);
    }
}

extern "C" void kernel_launch(void* const* d_in, const int* in_sizes, int n_in,
                              void* d_out, int out_size, void* d_ws, size_t ws_size,
                              hipStream_t stream) {
    (void)in_sizes; (void)n_in; (void)out_size; (void)d_ws; (void)ws_size;
    const float* x    = (const float*)d_in[0];
    const float* W    = (const float*)d_in[1];
    const float* bias = (const float*)d_in[2];
    float*       out  = (float*)d_out;
    winattn_wmma_kernel<<<dim3(S_), dim3(256), 0, stream>>>(x, W, bias, out);
}